// VectorQuantizerEMA_88149908783297
// MI455X (gfx1250) — compile-verified
//
#include <hip/hip_runtime.h>
#include <hip/hip_bf16.h>

typedef __attribute__((ext_vector_type(2))) float v2f;
typedef __attribute__((ext_vector_type(8))) float v8f;

static constexpr int Bn = 32, Dd = 256, Kk = 2048, HWn = 1024;
static constexpr int Nn = Bn * HWn;           // 32768
static constexpr int ASTR = 260;              // padded LDS row stride (dwords) -> bank-conflict-free
static constexpr float kBETA = 0.25f, kDECAY = 0.99f, kEPS = 1e-5f;

// ---------------------------------------------------------------- init
__global__ void init_zero(float* base, int n, float* out_loss) {
    int i = blockIdx.x * blockDim.x + threadIdx.x;
    if (i < n) base[i] = 0.0f;
    if (i == 0) out_loss[0] = 0.0f;
}

// ---------------------------------------------------------------- pack z_e [B,D,HW] -> ze [N,D]
__global__ void pack_ze(const float* __restrict__ z_e, float* __restrict__ ze) {
    __shared__ float tile[32][33];
    int tx = threadIdx.x, ty = threadIdx.y;
    int ht = blockIdx.x, dt = blockIdx.y, b = blockIdx.z;
    const float* src = z_e + (size_t)b * (Dd * HWn);
#pragma unroll
    for (int j = 0; j < 4; ++j) {
        int dl = ty * 4 + j;
        tile[dl][tx] = src[(dt * 32 + dl) * HWn + ht * 32 + tx];
    }
    __syncthreads();
    float* dst = ze + ((size_t)b * HWn + ht * 32) * Dd;
#pragma unroll
    for (int j = 0; j < 4; ++j) {
        int hl = ty * 4 + j;
        dst[(size_t)hl * Dd + dt * 32 + tx] = tile[tx][hl];
    }
}

// ---------------------------------------------------------------- embed [D,K] -> embedT [K,D]
__global__ void transpose_embed(const float* __restrict__ embed, float* __restrict__ embedT) {
    __shared__ float tile[32][33];
    int tx = threadIdx.x, ty = threadIdx.y;
    int kt = blockIdx.x, dt = blockIdx.y;
#pragma unroll
    for (int j = 0; j < 4; ++j) {
        int dl = ty * 4 + j;
        tile[dl][tx] = embed[(size_t)(dt * 32 + dl) * Kk + kt * 32 + tx];
    }
    __syncthreads();
#pragma unroll
    for (int j = 0; j < 4; ++j) {
        int kl = ty * 4 + j;
        embedT[(size_t)(kt * 32 + kl) * Dd + dt * 32 + tx] = tile[tx][kl];
    }
}

// ---------------------------------------------------------------- 0.5 * ||e_k||^2
__global__ void colnorm_half(const float* __restrict__ embed, float* __restrict__ cn) {
    int k = blockIdx.x * 256 + threadIdx.x;
    float s = 0.0f;
    for (int d = 0; d < Dd; ++d) {
        float e = embed[(size_t)d * Kk + k];
        s += e * e;
    }
    cn[k] = 0.5f * s;
}

// ---------------------------------------------------------------- main: fp32 WMMA GEMM + argmax
// Per block: 8 waves, 128 A-rows staged once in LDS; 128 K-tiles streamed with
// register-prefetch double buffering so the global B-tile latency hides behind
// the v_wmma_f32_16x16x4_f32 stream.
__global__ void __launch_bounds__(256)
vq_argmin_kernel(const float* __restrict__ ze,     // [N, 256]
                 const float* __restrict__ embed,  // [256, 2048]
                 const float* __restrict__ cn,     // [2048]
                 int* __restrict__ idx_ws,
                 float* __restrict__ out_idx,
                 float* __restrict__ counts) {
    extern __shared__ float smem[];
    float* As = smem;                  // [128][ASTR]
    float* Bs = smem + 128 * ASTR;     // [16][ASTR]  (transposed: Bs[col][d])

    const int tid  = threadIdx.x;
    const int wave = tid >> 5;
    const int lane = tid & 31;
    const int col  = lane & 15;
    const int half = lane >> 4;
    const int row0 = blockIdx.x * 128;

    // B-tile element this thread stages: Bs[bc][bd+i*16] = embed[(bd+i*16)*K + k0 + bc]
    const int bd = tid >> 4;           // 0..15  (d base; +16 per i)
    const int bc = tid & 15;           // 0..15  (column within tile)

    // cooperative A stage: 128 rows x 256 floats, float4 granularity
    {
        const float4* src = (const float4*)(ze + (size_t)row0 * Dd);
        for (int i = tid; i < 128 * 64; i += 256) {
            int r = i >> 6, c4 = i & 63;
            float4 v = src[r * 64 + c4];
            *(float4*)(As + r * ASTR + c4 * 4) = v;
        }
    }

    // prefetch B tile 0 into registers
    float breg[16];
#pragma unroll
    for (int i = 0; i < 16; ++i)
        breg[i] = embed[(size_t)(bd + i * 16) * Kk + bc];

    float best[8];
    int   bk[8];
#pragma unroll
    for (int r = 0; r < 8; ++r) { best[r] = -3.4e38f; bk[r] = 0; }

    const float* Arow = As + (wave * 16 + col) * ASTR + 2 * half;  // A: M = lane&15, K pair by half
    const float* Brow = Bs + col * ASTR + 2 * half;                // B: N = lane&15, K pair by half

    for (int kt = 0; kt < 128; ++kt) {
        const int k0 = kt << 4;
        __syncthreads();  // previous readers of Bs done (1st iter: As stage complete)
#pragma unroll
        for (int i = 0; i < 16; ++i)
            Bs[bc * ASTR + bd + i * 16] = breg[i];
        __syncthreads();

        // issue global loads for the next tile now; s_wait_loadcnt lands after the
        // next barrier, so the latency hides behind this iteration's WMMA work
        if (kt < 127) {
            const int k0n = k0 + 16;
#pragma unroll
            for (int i = 0; i < 16; ++i)
                breg[i] = embed[(size_t)(bd + i * 16) * Kk + k0n + bc];
        }

        v8f acc0 = {0.f, 0.f, 0.f, 0.f, 0.f, 0.f, 0.f, 0.f};
        v8f acc1 = {0.f, 0.f, 0.f, 0.f, 0.f, 0.f, 0.f, 0.f};
#pragma unroll 8
        for (int d8 = 0; d8 < 32; ++d8) {
            v2f a0 = *(const v2f*)(Arow + (d8 << 3));
            v2f b0 = *(const v2f*)(Brow + (d8 << 3));
            acc0 = __builtin_amdgcn_wmma_f32_16x16x4_f32(false, a0, false, b0,
                                                         (short)0, acc0, false, false);
            v2f a1 = *(const v2f*)(Arow + (d8 << 3) + 4);
            v2f b1 = *(const v2f*)(Brow + (d8 << 3) + 4);
            acc1 = __builtin_amdgcn_wmma_f32_16x16x4_f32(false, a1, false, b1,
                                                         (short)0, acc1, false, false);
        }

        const float ch  = cn[k0 + col];
        const int   myk = k0 + col;
#pragma unroll
        for (int r = 0; r < 8; ++r) {
            float s = acc0[r] + acc1[r] - ch;   // score = dot - 0.5||e||^2 (argmax == argmin dist)
            if (s > best[r]) { best[r] = s; bk[r] = myk; }  // strict > keeps smallest k on ties
        }
    }

    // argmax reduce across the 16 lanes of each half (tie -> smaller k)
#pragma unroll
    for (int m = 1; m < 16; m <<= 1) {
#pragma unroll
        for (int r = 0; r < 8; ++r) {
            float ov = __shfl_xor(best[r], m, 32);
            int   ok = __shfl_xor(bk[r], m, 32);
            if (ov > best[r] || (ov == best[r] && ok < bk[r])) { best[r] = ov; bk[r] = ok; }
        }
    }

    if (col == 0) {
#pragma unroll
        for (int r = 0; r < 8; ++r) {
            int grow = row0 + wave * 16 + half * 8 + r;  // D-matrix: M = vgpr + 8*half
            idx_ws[grow]  = bk[r];
            out_idx[grow] = (float)bk[r];
            atomicAdd(&counts[bk[r]], 1.0f);
        }
    }
}

// ---------------------------------------------------------------- gather z_q, scatter embed_sum, loss
__global__ void zq_scatter(const float* __restrict__ ze, const float* __restrict__ embedT,
                           const int* __restrict__ idx, float* __restrict__ out0,
                           float* __restrict__ embed_sum, float* __restrict__ loss_part) {
    __shared__ float t2[32][33];
    __shared__ int ks[32];
    __shared__ float wsum[8];
    int tx = threadIdx.x, ty = threadIdx.y;
    int n0 = blockIdx.x * 32, d0 = blockIdx.y * 32;
    int b = n0 >> 10, hw0 = n0 & (HWn - 1);
    int t = ty * 32 + tx;
    if (ty == 0) ks[tx] = idx[n0 + tx];
    __syncthreads();
    float lp = 0.0f;
#pragma unroll
    for (int j = 0; j < 4; ++j) {
        int nl = ty * 4 + j;
        int k = ks[nl];
        float q = embedT[(size_t)k * Dd + d0 + tx];
        float z = ze[(size_t)(n0 + nl) * Dd + d0 + tx];
        t2[nl][tx] = q;
        float df = z - q;
        lp += df * df;
        atomicAdd(&embed_sum[(size_t)(d0 + tx) * Kk + k], z);
    }
    __syncthreads();
#pragma unroll
    for (int j = 0; j < 4; ++j) {
        int dl = ty * 4 + j;
        out0[(size_t)b * (Dd * HWn) + (size_t)(d0 + dl) * HWn + hw0 + tx] = t2[tx][dl];
    }
    for (int m = 16; m > 0; m >>= 1) lp += __shfl_xor(lp, m, 32);
    int wave = t >> 5, lane = t & 31;
    if (lane == 0) wsum[wave] = lp;
    __syncthreads();
    if (t == 0) {
        float s = 0.0f;
#pragma unroll
        for (int w = 0; w < 8; ++w) s += wsum[w];
        atomicAdd(&loss_part[blockIdx.x & 1023], s);
    }
}

// ---------------------------------------------------------------- EMA stage A: cluster size, n, loss
__global__ void ema_a(const float* __restrict__ cs_in, const float* __restrict__ counts,
                      const float* __restrict__ loss_part, float* __restrict__ out_ncs,
                      float* __restrict__ out_loss, float* __restrict__ nsum) {
    __shared__ float r1[256], r2[256];
    int t = threadIdx.x;
    float la = 0.0f;
    for (int i = t; i < 1024; i += 256) la += loss_part[i];
    float na = 0.0f;
    for (int k = t; k < Kk; k += 256) {
        float v = cs_in[k] * kDECAY + (1.0f - kDECAY) * counts[k];
        out_ncs[k] = v;
        na += v;
    }
    r1[t] = la; r2[t] = na;
    __syncthreads();
    for (int s = 128; s > 0; s >>= 1) {
        if (t < s) { r1[t] += r1[t + s]; r2[t] += r2[t + s]; }
        __syncthreads();
    }
    if (t == 0) {
        out_loss[0] = r1[0] * (kBETA / (float)(Nn * Dd));
        nsum[0] = r2[0];
    }
}

// ---------------------------------------------------------------- EMA stage B: embed update
__global__ void ema_b(const float* __restrict__ embed_avg, const float* __restrict__ embed_sum,
                      const float* __restrict__ out_ncs, const float* __restrict__ nsum,
                      float* __restrict__ out_embed, float* __restrict__ out_embed_avg) {
    int i = blockIdx.x * blockDim.x + threadIdx.x;  // 0 .. D*K-1, k fastest
    int k = i & (Kk - 1);
    float es = embed_avg[i] * kDECAY + (1.0f - kDECAY) * embed_sum[i];
    out_embed_avg[i] = es;
    float n = nsum[0];
    float ncs = out_ncs[k];
    float cs = (ncs + kEPS) / (n + (float)Kk * kEPS) * n;
    out_embed[i] = es / cs;
}

// ---------------------------------------------------------------- launch
extern "C" void kernel_launch(void* const* d_in, const int* in_sizes, int n_in,
                              void* d_out, int out_size, void* d_ws, size_t ws_size,
                              hipStream_t stream) {
    const float* z_e          = (const float*)d_in[0];  // [32,256,32,32]
    const float* embed        = (const float*)d_in[1];  // [256,2048]
    const float* cluster_size = (const float*)d_in[2];  // [2048]
    const float* embed_avg    = (const float*)d_in[3];  // [256,2048]

    float* out  = (float*)d_out;
    float* out0 = out;                       // z_q_st       [32,256,32,32]
    float* out1 = out0 + (size_t)Nn * Dd;    // idx (float)  [32768]
    float* out2 = out1 + Nn;                 // commit_loss  [1]
    float* out3 = out2 + 1;                  // new_embed    [256,2048]
    float* out4 = out3 + (size_t)Dd * Kk;    // new_cluster  [2048]
    float* out5 = out4 + Kk;                 // new_embed_avg[256,2048]

    float* ws        = (float*)d_ws;
    float* ze_ws     = ws;                                  // 8388608
    float* embedT    = ze_ws + (size_t)Nn * Dd;             // 524288
    float* cn        = embedT + (size_t)Kk * Dd;            // 2048
    int*   idx_ws    = (int*)(cn + Kk);                     // 32768
    float* embed_sum = (float*)(idx_ws + Nn);               // 524288  (zeroed region start)
    float* counts    = embed_sum + (size_t)Dd * Kk;         // 2048
    float* nsum      = counts + Kk;                         // 1
    float* loss_part = nsum + 1;                            // 1024
    const int zero_n = Dd * Kk + Kk + 1 + 1024;             // contiguous accumulators

    // 0) zero accumulators + loss output
    init_zero<<<(zero_n + 255) / 256, 256, 0, stream>>>(embed_sum, zero_n, out2);
    // 1) pack z_e -> ze [N,D]
    pack_ze<<<dim3(32, 8, 32), dim3(32, 8), 0, stream>>>(z_e, ze_ws);
    // 2) embedT [K,D]
    transpose_embed<<<dim3(64, 8), dim3(32, 8), 0, stream>>>(embed, embedT);
    // 3) 0.5*colnorms
    colnorm_half<<<Kk / 256, 256, 0, stream>>>(embed, cn);
    // 4) fp32 WMMA GEMM + argmin
    hipFuncSetAttribute((const void*)vq_argmin_kernel,
                        hipFuncAttributeMaxDynamicSharedMemorySize, (128 + 16) * ASTR * 4);
    vq_argmin_kernel<<<Nn / 128, 256, (128 + 16) * ASTR * 4, stream>>>(
        ze_ws, embed, cn, idx_ws, out1, counts);
    // 5) z_q gather/transpose + embed_sum scatter + loss partials
    zq_scatter<<<dim3(Nn / 32, Dd / 32), dim3(32, 8), 0, stream>>>(
        ze_ws, embedT, idx_ws, out0, embed_sum, loss_part);
    // 6) cluster-size EMA + n + loss
    ema_a<<<1, 256, 0, stream>>>(cluster_size, counts, loss_part, out4, out2, nsum);
    // 7) embed update
    ema_b<<<(Dd * Kk) / 256, 256, 0, stream>>>(embed_avg, embed_sum, out4, nsum, out3, out5);
}